// DisentangledMHA_69535520522847
// MI455X (gfx1250) — compile-verified
//
#include <hip/hip_runtime.h>
#include <hip/hip_bf16.h>

// DeBERTa disentangled MHA for gfx1250 (MI455X), fp32 WMMA path.
// scores[q,k] = (q.k + q.pos_k[q-k+512] + k.pos_q[q-k+512]) * scale + mask
// (clip is a no-op for s == MAX_REL == 512).

typedef __attribute__((ext_vector_type(2))) float v2f;
typedef __attribute__((ext_vector_type(8))) float v8f;

#define DIM   1024
#define SEQ   512
#define NHEAD 16
#define HDIM  64
#define SROW  520                 // padded LDS stride (floats) for score rows
#define GBUF  512                 // 16x32 / 32x16 gather staging (floats)
#define WAVE_LDS (16 * SROW + GBUF)
#define SCORE_SCALE 0.07216878364870322f   // 1/sqrt(64*3)

__device__ __forceinline__ v8f wmma4(v2f a, v2f b, v8f c) {
  // V_WMMA_F32_16X16X4_F32 : D = A(16x4) x B(4x16) + C(16x16)
  return __builtin_amdgcn_wmma_f32_16x16x4_f32(
      false, a, false, b, (short)0, c, false, false);
}

__device__ __forceinline__ int clampi(int v, int lo, int hi) {
  return v < lo ? lo : (v > hi ? hi : v);
}

// ---------------------------------------------------------------------------
// Y[M,1024] = X[M,1024] @ W[1024,1024]^T + bias    (one wave per 16x64 tile)
// ---------------------------------------------------------------------------
__global__ void gemm_bias_nt(const float* __restrict__ X,
                             const float* __restrict__ W,
                             const float* __restrict__ bias,
                             float* __restrict__ Y, int M) {
  const int lane = threadIdx.x & 31;
  const int gw   = blockIdx.x * (blockDim.x >> 5) + (threadIdx.x >> 5);
  const int ntile = DIM / 64;              // 16 N-tiles of 64
  const int tM = gw / ntile;
  const int tN = gw % ntile;
  if (tM >= (M >> 4)) return;              // wave-uniform guard
  const int l15 = lane & 15;
  const int kh  = (lane >> 4) << 1;        // K sub-offset 0 / 2
  const int mh  = (lane >> 4) << 3;        // D-frag row offset 0 / 8
  const int row = tM * 16 + l15;

  v8f acc[4] = {};
  const float* xrow = X + (size_t)row * DIM;
  for (int k0 = 0; k0 < DIM; k0 += 4) {
    v2f a = *(const v2f*)(xrow + k0 + kh);             // A-frag of X
#pragma unroll
    for (int nd = 0; nd < 4; ++nd) {
      const int wr = tN * 64 + nd * 16 + l15;          // B-frag of W^T
      v2f b = *(const v2f*)(W + (size_t)wr * DIM + k0 + kh);
      acc[nd] = wmma4(a, b, acc[nd]);
    }
  }
#pragma unroll
  for (int nd = 0; nd < 4; ++nd) {
    const int col = tN * 64 + nd * 16 + l15;
    const float bv = bias[col];
#pragma unroll
    for (int r = 0; r < 8; ++r) {
      const int orow = tM * 16 + r + mh;
      Y[(size_t)orow * DIM + col] = acc[nd][r] + bv;
    }
  }
}

// ---------------------------------------------------------------------------
// Attention: one wave per (b, h, 16-row q tile).  Q/K/V are [4096,1024]
// row-major ([b,s,h,d]); PQ/PK are [1024,1024] ([2s,h,d]).
// ---------------------------------------------------------------------------
__global__ void attn_kernel(const float* __restrict__ Qm,
                            const float* __restrict__ Km,
                            const float* __restrict__ Vm,
                            const float* __restrict__ PQ,
                            const float* __restrict__ PK,
                            const float* __restrict__ mask,
                            float* __restrict__ out) {
  extern __shared__ float lds[];
  const int lane = threadIdx.x & 31;
  const int wid  = threadIdx.x >> 5;
  const int gw   = blockIdx.x * (blockDim.x >> 5) + wid;   // 0..4095
  const int qt = gw & 31;
  const int h  = (gw >> 5) & 15;
  const int bi = gw >> 9;
  const int Q0 = qt * 16;
  const int hc = h * HDIM;

  float* sc   = lds + wid * WAVE_LDS;   // 16 x SROW score block (wave private)
  float* gbuf = sc + 16 * SROW;         // 512-float gather staging

  const int l15 = lane & 15;
  const int kh  = (lane >> 4) << 1;     // K sub-offset 0/2
  const int mh  = (lane >> 4) << 3;     // D-frag row offset 0/8
  const int rowQ = bi * SEQ + Q0 + l15;

  // Preload this tile's q A-fragments (d = 0..63 in 16 k-steps of 4).
  v2f qa[16];
#pragma unroll
  for (int ks = 0; ks < 16; ++ks)
    qa[ks] = *(const v2f*)(Qm + (size_t)rowQ * DIM + hc + ks * 4 + kh);

  // ---- Pass 1: scores -> LDS ----
  for (int kt = 0; kt < 32; ++kt) {
    const int K0 = kt * 16;
    const int rowK = bi * SEQ + K0 + l15;
    const int J0 = Q0 - K0 + SEQ - 15;   // pos row of band start

    // q . k
    v8f s = {};
#pragma unroll
    for (int ks = 0; ks < 16; ++ks) {
      v2f b = *(const v2f*)(Km + (size_t)rowK * DIM + hc + ks * 4 + kh);
      s = wmma4(qa[ks], b, s);
    }

    // c2p band: T[qi][c] = q[Q0+qi] . pos_k[J0+c],  c in [0,32)
    v8f t0 = {}, t1 = {};
    {
      const int pr0 = clampi(J0 + l15, 0, 2 * SEQ - 1);
      const int pr1 = clampi(J0 + 16 + l15, 0, 2 * SEQ - 1);
#pragma unroll
      for (int ks = 0; ks < 16; ++ks) {
        v2f b0 = *(const v2f*)(PK + (size_t)pr0 * DIM + hc + ks * 4 + kh);
        v2f b1 = *(const v2f*)(PK + (size_t)pr1 * DIM + hc + ks * 4 + kh);
        t0 = wmma4(qa[ks], b0, t0);
        t1 = wmma4(qa[ks], b1, t1);
      }
    }
    // stage [qi][c] (16x32) and gather diagonal c = 15 + qi - kj
#pragma unroll
    for (int r = 0; r < 8; ++r) {
      gbuf[(r + mh) * 32 + l15]      = t0[r];
      gbuf[(r + mh) * 32 + 16 + l15] = t1[r];
    }
    asm volatile("s_wait_dscnt 0" ::: "memory");
#pragma unroll
    for (int r = 0; r < 8; ++r) {
      const int qi = r + mh;
      s[r] += gbuf[qi * 32 + (15 + qi - l15)];
    }
    asm volatile("" ::: "memory");

    // p2c band (transposed): U[jr][kj] = pos_q[J0+jr] . k[K0+kj]
    v8f u0 = {}, u1 = {};
    {
      const int pr0 = clampi(J0 + l15, 0, 2 * SEQ - 1);
      const int pr1 = clampi(J0 + 16 + l15, 0, 2 * SEQ - 1);
#pragma unroll
      for (int ks = 0; ks < 16; ++ks) {
        v2f a0 = *(const v2f*)(PQ + (size_t)pr0 * DIM + hc + ks * 4 + kh);
        v2f a1 = *(const v2f*)(PQ + (size_t)pr1 * DIM + hc + ks * 4 + kh);
        v2f b  = *(const v2f*)(Km + (size_t)rowK * DIM + hc + ks * 4 + kh);
        u0 = wmma4(a0, b, u0);
        u1 = wmma4(a1, b, u1);
      }
    }
    // stage [jr][kj] (32x16) and gather jr = 15 + qi - kj
#pragma unroll
    for (int r = 0; r < 8; ++r) {
      gbuf[(r + mh) * 16 + l15]        = u0[r];
      gbuf[(r + mh + 16) * 16 + l15]   = u1[r];
    }
    asm volatile("s_wait_dscnt 0" ::: "memory");
    const float mv = mask[bi * SEQ + K0 + l15];
#pragma unroll
    for (int r = 0; r < 8; ++r) {
      const int qi = r + mh;
      float v = s[r] + gbuf[(15 + qi - l15) * 16 + l15];
      sc[qi * SROW + K0 + l15] = v * SCORE_SCALE + mv;
    }
    asm volatile("" ::: "memory");
  }

  // ---- Softmax over 512 columns, rows 0..15 (wave cooperative) ----
  for (int r = 0; r < 16; ++r) {
    float* row = sc + r * SROW;
    float m = -3.4e38f;
    for (int c = lane; c < SEQ; c += 32) m = fmaxf(m, row[c]);
#pragma unroll
    for (int off = 16; off > 0; off >>= 1) m = fmaxf(m, __shfl_xor(m, off, 32));
    float sum = 0.f;
    for (int c = lane; c < SEQ; c += 32) {
      float e = __expf(row[c] - m);
      row[c] = e;
      sum += e;
    }
#pragma unroll
    for (int off = 16; off > 0; off >>= 1) sum += __shfl_xor(sum, off, 32);
    const float inv = 1.f / sum;
    for (int c = lane; c < SEQ; c += 32) row[c] *= inv;
  }
  asm volatile("s_wait_dscnt 0" ::: "memory");

  // ---- P @ V : out tile 16 x 64 ----
  v8f o[4] = {};
  for (int kt = 0; kt < 32; ++kt) {
    const int K0 = kt * 16;
#pragma unroll
    for (int ks = 0; ks < 4; ++ks) {
      // A-frag of probs straight from LDS (row qi = l15, 2 consecutive cols)
      v2f a = *(const v2f*)(sc + l15 * SROW + K0 + ks * 4 + kh);
      const int vr = bi * SEQ + K0 + ks * 4 + kh;       // B rows of V (K dim)
#pragma unroll
      for (int nd = 0; nd < 4; ++nd) {
        const int col = hc + nd * 16 + l15;
        v2f b;
        b[0] = Vm[(size_t)vr * DIM + col];
        b[1] = Vm[(size_t)(vr + 1) * DIM + col];
        o[nd] = wmma4(a, b, o[nd]);
      }
    }
  }

  // out[b, s, h*64+d]
#pragma unroll
  for (int nd = 0; nd < 4; ++nd) {
#pragma unroll
    for (int r = 0; r < 8; ++r) {
      const int qi = Q0 + r + mh;
      out[((size_t)(bi * SEQ + qi)) * DIM + hc + nd * 16 + l15] = o[nd][r];
    }
  }
}

// ---------------------------------------------------------------------------
extern "C" void kernel_launch(void* const* d_in, const int* in_sizes, int n_in,
                              void* d_out, int out_size, void* d_ws, size_t ws_size,
                              hipStream_t stream) {
  (void)in_sizes; (void)n_in; (void)out_size; (void)ws_size;
  const float* x    = (const float*)d_in[0];
  const float* rel  = (const float*)d_in[1];   // [1024,1024] (full att_span slice)
  const float* mask = (const float*)d_in[2];
  const float* Wq   = (const float*)d_in[3];
  const float* bq   = (const float*)d_in[4];
  const float* Wk   = (const float*)d_in[5];
  const float* bk   = (const float*)d_in[6];
  const float* Wv   = (const float*)d_in[7];
  const float* bv   = (const float*)d_in[8];
  float* out = (float*)d_out;

  float* Qm = (float*)d_ws;                       // 4096x1024
  float* Km = Qm + (size_t)4096 * 1024;           // 4096x1024
  float* Vm = Km + (size_t)4096 * 1024;           // 4096x1024
  float* PQ = Vm + (size_t)4096 * 1024;           // 1024x1024
  float* PK = PQ + (size_t)1024 * 1024;           // 1024x1024

  // Projections: 16x64 tile per wave, 4 waves per block.
  gemm_bias_nt<<<1024, 128, 0, stream>>>(x,   Wq, bq, Qm, 4096);
  gemm_bias_nt<<<1024, 128, 0, stream>>>(x,   Wk, bk, Km, 4096);
  gemm_bias_nt<<<1024, 128, 0, stream>>>(x,   Wv, bv, Vm, 4096);
  gemm_bias_nt<<<256,  128, 0, stream>>>(rel, Wq, bq, PQ, 1024);
  gemm_bias_nt<<<256,  128, 0, stream>>>(rel, Wk, bk, PK, 1024);

  // Attention: 4096 waves (8 b x 16 h x 32 q-tiles), 2 waves/block.
  const size_t shmem = (size_t)2 * WAVE_LDS * sizeof(float);  // ~70.7 KB
  attn_kernel<<<2048, 64, shmem, stream>>>(Qm, Km, Vm, PQ, PK, mask, out);
}